// GraphModel_43636867727979
// MI455X (gfx1250) — compile-verified
//
#include <hip/hip_runtime.h>
#include <hip/hip_bf16.h>

// ---------------------------------------------------------------------------
// GraphModel on MI455X (gfx1250, wave32).
//   - Dense GEMMs via V_WMMA_F32_16X16X4_F32, 16x64 output tile per wave
//     (A-fragment reused across 4 WMMAs -> 2.5 loads/WMMA instead of 6).
//   - Graph aggregation done structurally (window +/-6, cross-modal, self),
//     closed-form degrees, float4 (b128) loads; no atomics.
// ---------------------------------------------------------------------------

typedef __attribute__((ext_vector_type(2))) float v2f;
typedef __attribute__((ext_vector_type(8))) float v8f;

#define GB  64      // dialogues
#define GL  128     // utterances per dialogue
#define GN  8192    // GB*GL
#define GNN 32768   // 4*GN
#define ND  512     // input dim
#define NH  256     // hidden dim

// ---------------------------------------------------------------------------
// spk_idx: argmax over 2 speakers. qmask is [L, B, 2]; node i = b*L + t.
// ---------------------------------------------------------------------------
__global__ void spk_kernel(const float* __restrict__ qmask,
                           int* __restrict__ spk_ws,
                           int* __restrict__ spk_out) {
  int i = blockIdx.x * blockDim.x + threadIdx.x;
  if (i >= GN) return;
  int b = i >> 7;            // i / L
  int t = i & (GL - 1);      // i % L
  float q0 = qmask[(t * GB + b) * 2 + 0];
  float q1 = qmask[(t * GB + b) * 2 + 1];
  int idx = (q1 > q0) ? 1 : 0;   // jnp.argmax: first index on tie
  spk_ws[i]  = idx;
  spk_out[i] = idx;
}

// ---------------------------------------------------------------------------
// WMMA fragment helpers.
// A 16x4 layout: lane half h supplies K = 2h (v0) and K = 2h+1 (v1) of row l16.
// B 4x16 mirrored. C/D: VGPR g -> row g + 8*half, col l16.
// ---------------------------------------------------------------------------
__device__ __forceinline__ v8f wmma4(v2f a, v2f b, v8f c) {
  return __builtin_amdgcn_wmma_f32_16x16x4_f32(false, a, false, b,
                                               (short)0, c, false, false);
}

// ---------------------------------------------------------------------------
// GEMM stage 1: out[NN,256] = (X_m + spk_emb) @ W[512,256] + bias
// One wave per 16x64 output tile; feats built on the fly (never materialized).
// ---------------------------------------------------------------------------
__global__ __launch_bounds__(256) void gemm_feats_kernel(
    const float* __restrict__ a, const float* __restrict__ v,
    const float* __restrict__ l, const float* __restrict__ u,
    const float* __restrict__ spkT, const int* __restrict__ spk_idx,
    const float* __restrict__ W, const float* __restrict__ bias,
    float* __restrict__ out) {
  int wave = (blockIdx.x * blockDim.x + threadIdx.x) >> 5;
  int lane = threadIdx.x & 31;
  int tn4 = wave & 3;          // 4 col-tiles of 64 (256/64)
  int tm  = wave >> 2;         // 2048 row tiles (32768/16)
  int half = lane >> 4;
  int l16  = lane & 15;

  int row = tm * 16 + l16;     // tile never crosses modality boundary
  int m = row >> 13;           // row / 8192
  int i = row & (GN - 1);
  const float* X = (m == 0) ? a : (m == 1) ? v : (m == 2) ? l : u;
  const float* Arow = X + (size_t)i * ND;
  const float* Srow = spkT + (size_t)spk_idx[i] * ND;
  int col0 = tn4 * 64 + l16;   // this lane's column in sub-tile 0

  v8f c0 = {}, c1 = {}, c2 = {}, c3 = {};
  #pragma unroll 2
  for (int k = 0; k < ND; k += 4) {
    int kk = k + 2 * half;
    v2f af;
    af.x = Arow[kk]     + Srow[kk];
    af.y = Arow[kk + 1] + Srow[kk + 1];
    const float* W0 = W + (size_t)kk * NH;        // row kk
    const float* W1 = W0 + NH;                    // row kk+1
    v2f b0 = { W0[col0],      W1[col0]      };
    v2f b1 = { W0[col0 + 16], W1[col0 + 16] };
    v2f b2 = { W0[col0 + 32], W1[col0 + 32] };
    v2f b3 = { W0[col0 + 48], W1[col0 + 48] };
    c0 = wmma4(af, b0, c0);
    c1 = wmma4(af, b1, c1);
    c2 = wmma4(af, b2, c2);
    c3 = wmma4(af, b3, c3);
  }
  float* outp = out + (size_t)(tm * 16 + 8 * half) * NH + col0;
  float bv0 = bias[col0],      bv1 = bias[col0 + 16];
  float bv2 = bias[col0 + 32], bv3 = bias[col0 + 48];
  #pragma unroll
  for (int g = 0; g < 8; ++g) {
    outp[(size_t)g * NH +  0] = c0[g] + bv0;
    outp[(size_t)g * NH + 16] = c1[g] + bv1;
    outp[(size_t)g * NH + 32] = c2[g] + bv2;
    outp[(size_t)g * NH + 48] = c3[g] + bv3;
  }
}

// ---------------------------------------------------------------------------
// GEMM stage 2: out[NN,256] = x[NN,256] @ W[256,256] + bias, 16x64 tiles.
// ---------------------------------------------------------------------------
__global__ __launch_bounds__(256) void gemm_plain_kernel(
    const float* __restrict__ x, const float* __restrict__ W,
    const float* __restrict__ bias, float* __restrict__ out) {
  int wave = (blockIdx.x * blockDim.x + threadIdx.x) >> 5;
  int lane = threadIdx.x & 31;
  int tn4 = wave & 3;
  int tm  = wave >> 2;
  int half = lane >> 4;
  int l16  = lane & 15;

  const float* Arow = x + (size_t)(tm * 16 + l16) * NH;
  int col0 = tn4 * 64 + l16;

  v8f c0 = {}, c1 = {}, c2 = {}, c3 = {};
  #pragma unroll 2
  for (int k = 0; k < NH; k += 4) {
    int kk = k + 2 * half;
    v2f af = { Arow[kk], Arow[kk + 1] };
    const float* W0 = W + (size_t)kk * NH;
    const float* W1 = W0 + NH;
    v2f b0 = { W0[col0],      W1[col0]      };
    v2f b1 = { W0[col0 + 16], W1[col0 + 16] };
    v2f b2 = { W0[col0 + 32], W1[col0 + 32] };
    v2f b3 = { W0[col0 + 48], W1[col0 + 48] };
    c0 = wmma4(af, b0, c0);
    c1 = wmma4(af, b1, c1);
    c2 = wmma4(af, b2, c2);
    c3 = wmma4(af, b3, c3);
  }
  float* outp = out + (size_t)(tm * 16 + 8 * half) * NH + col0;
  float bv0 = bias[col0],      bv1 = bias[col0 + 16];
  float bv2 = bias[col0 + 32], bv3 = bias[col0 + 48];
  #pragma unroll
  for (int g = 0; g < 8; ++g) {
    outp[(size_t)g * NH +  0] = c0[g] + bv0;
    outp[(size_t)g * NH + 16] = c1[g] + bv1;
    outp[(size_t)g * NH + 32] = c2[g] + bv2;
    outp[(size_t)g * NH + 48] = c3[g] + bv3;
  }
}

// ---------------------------------------------------------------------------
// Structural graph aggregation (float4 / b128 loads):
//   agg[node] = mean over { self, t+/-1..6 (same modality, same dialogue),
//                           cross-modal u<->{a,v,l} }
//   out = relu(agg) + eps * init ; optional fused write of u-slice into d_out.
// Thread t handles 4 channels; 64 threads per node, 4 nodes per block.
// ---------------------------------------------------------------------------
__device__ __forceinline__ float4 f4add(float4 x, float4 y) {
  return make_float4(x.x + y.x, x.y + y.y, x.z + y.z, x.w + y.w);
}

__global__ __launch_bounds__(256) void agg_kernel(
    const float4* __restrict__ h4, const float4* __restrict__ init4,
    float4* __restrict__ out4, float4* __restrict__ out1_4,
    int col_off4, float eps) {
  int tid  = blockIdx.x * blockDim.x + threadIdx.x;   // over GNN*64
  int node = tid >> 6;
  int q    = tid & 63;          // float4 index within row (NH/4 = 64)
  int m = node >> 13;           // node / 8192
  int i = node & (GN - 1);
  int t = i & (GL - 1);         // position in dialogue

  float4 s = h4[node * 64 + q]; // self loop
  int lo = (t < 6) ? t : 6;
  int hi = ((GL - 1 - t) < 6) ? (GL - 1 - t) : 6;
  for (int o = 1; o <= lo; ++o) s = f4add(s, h4[(node - o) * 64 + q]);
  for (int o = 1; o <= hi; ++o) s = f4add(s, h4[(node + o) * 64 + q]);

  float deg;
  if (m < 3) {                  // a/v/l receive from u
    s = f4add(s, h4[(3 * GN + i) * 64 + q]);
    deg = (float)(lo + hi + 2);
  } else {                      // u receives from a, v, l
    s = f4add(s, h4[i * 64 + q]);
    s = f4add(s, h4[(GN + i) * 64 + q]);
    s = f4add(s, h4[(2 * GN + i) * 64 + q]);
    deg = (float)(lo + hi + 4);
  }

  float rdeg = 1.0f / deg;
  float4 ini = init4[node * 64 + q];
  float4 val = make_float4(fmaxf(s.x * rdeg, 0.0f) + eps * ini.x,
                           fmaxf(s.y * rdeg, 0.0f) + eps * ini.y,
                           fmaxf(s.z * rdeg, 0.0f) + eps * ini.z,
                           fmaxf(s.w * rdeg, 0.0f) + eps * ini.w);
  out4[node * 64 + q] = val;
  if (out1_4 != nullptr && m == 3)   // fused final-layer u-slice write
    out1_4[(size_t)i * 128 + col_off4 + q] = val;  // d_out row = 512 f = 128 f4
}

// ---------------------------------------------------------------------------
extern "C" void kernel_launch(void* const* d_in, const int* in_sizes, int n_in,
                              void* d_out, int out_size, void* d_ws, size_t ws_size,
                              hipStream_t stream) {
  const float* a     = (const float*)d_in[0];
  const float* v     = (const float*)d_in[1];
  const float* l     = (const float*)d_in[2];
  const float* u     = (const float*)d_in[3];
  const float* qmask = (const float*)d_in[4];
  const float* spkT  = (const float*)d_in[5];
  const float* fc1_W = (const float*)d_in[6];
  const float* fc1_b = (const float*)d_in[7];
  const float* fc2_W = (const float*)d_in[8];
  const float* fc2_b = (const float*)d_in[9];
  const float* Wk    = (const float*)d_in[10];  // [4,256,256]
  const float* bk    = (const float*)d_in[11];  // [4,256]
  const float* Wl    = (const float*)d_in[12];  // [3,256,256]
  const float* bl    = (const float*)d_in[13];  // [3,256]
  // edge arrays d_in[14..17] are deterministic (_build_edges) -> computed structurally

  float4* out1_4 = (float4*)d_out;                           // [8192, 512] f32
  int*    spk_out = (int*)((float*)d_out + (size_t)GN * 512); // spk_idx tail

  // Workspace: spk_idx (64KB slot) + three 32MB activation buffers.
  const size_t SZ = (size_t)GNN * NH;   // elements per activation matrix
  char*  ws  = (char*)d_ws;
  int*   spk = (int*)ws;
  float* B1  = (float*)(ws + 65536);    // init / residual source
  float* B2  = B1 + SZ;                 // conv output (ping)
  float* B3  = B2 + SZ;                 // GEMM result h

  const int GEMM_BLOCKS = (2048 * 4) / 8;   // 8192 waves / 8 waves-per-block
  const int AGG_BLOCKS  = (GNN * 64) / 256; // 8192 blocks

  // speaker indices
  spk_kernel<<<GN / 256, 256, 0, stream>>>(qmask, spk, spk_out);

  // ---- ECG branch: gnn0 = feats @ fc2 ; 4 conv layers, eps1 = 1.0 ----
  gemm_feats_kernel<<<GEMM_BLOCKS, 256, 0, stream>>>(a, v, l, u, spkT, spk,
                                                     fc2_W, fc2_b, B1);
  {
    const float* cur = B1;
    for (int k = 0; k < 4; ++k) {
      gemm_plain_kernel<<<GEMM_BLOCKS, 256, 0, stream>>>(
          cur, Wk + (size_t)k * NH * NH, bk + (size_t)k * NH, B3);
      agg_kernel<<<AGG_BLOCKS, 256, 0, stream>>>(
          (const float4*)B3, (const float4*)B1, (float4*)B2,
          (k == 3) ? out1_4 : nullptr, NH / 4, 1.0f);
      cur = B2;
    }
  }

  // ---- EIG branch: x1 = feats @ fc1 ; 3 conv layers, eps2 = 1.0 ----
  gemm_feats_kernel<<<GEMM_BLOCKS, 256, 0, stream>>>(a, v, l, u, spkT, spk,
                                                     fc1_W, fc1_b, B1);
  {
    const float* cur = B1;
    for (int ll = 0; ll < 3; ++ll) {
      gemm_plain_kernel<<<GEMM_BLOCKS, 256, 0, stream>>>(
          cur, Wl + (size_t)ll * NH * NH, bl + (size_t)ll * NH, B3);
      agg_kernel<<<AGG_BLOCKS, 256, 0, stream>>>(
          (const float4*)B3, (const float4*)B1, (float4*)B2,
          (ll == 2) ? out1_4 : nullptr, 0, 1.0f);
      cur = B2;
    }
  }
}